// BatchedMLPHeads_67027259621900
// MI455X (gfx1250) — compile-verified
//
#include <hip/hip_runtime.h>

typedef __attribute__((ext_vector_type(2))) float v2f;
typedef __attribute__((ext_vector_type(8))) float v8f;

#define N_ASSETS 1024
#define D_MODEL  256
#define HIDDEN   1024
#define BATCH    2048
#define HPAD     260   // padded LDS row stride (floats): 260 % 64 banks = 4 -> conflict-free B reads

// One block per asset. Block = 256 threads = 8 wave32.
// Phase 1: scan asset_ids, collect this asset's sample indices (LDS atomic append).
// Phase 2: per chunk of <=16 samples: stage h columns in LDS, then each wave
//          computes 8 tiles of 16 hidden rows via v_wmma_f32_16x16x4_f32
//          (A = W1[a] 16x4 tile from global, B = 4x16 of h columns from LDS),
//          applies bias+relu+W2 dot in the D-fragment layout, reduces across waves.
__global__ __launch_bounds__(256)
void asset_mlp_wmma_kernel(const float* __restrict__ h,
                           const int*   __restrict__ ids,
                           const float* __restrict__ W1,
                           const float* __restrict__ b1,
                           const float* __restrict__ W2,
                           const float* __restrict__ b2,
                           float* __restrict__ out)
{
    __shared__ int   s_idx[BATCH];
    __shared__ int   s_count;
    __shared__ float s_h[16][HPAD];
    __shared__ float s_red[8][32];

    const int asset = blockIdx.x;
    const int tid   = threadIdx.x;
    const int wave  = tid >> 5;
    const int lane  = tid & 31;

    if (tid == 0) s_count = 0;
    __syncthreads();

    // ---- Phase 1: find samples belonging to this asset ----
    for (int i = tid; i < BATCH; i += 256) {
        if (ids[i] == asset) {
            int p = atomicAdd(&s_count, 1);
            s_idx[p] = i;
        }
    }
    __syncthreads();
    const int cnt = s_count;
    if (cnt == 0) return;   // block-uniform: skip reading W1[a] entirely

    const float* __restrict__ W1a = W1 + (size_t)asset * HIDDEN * D_MODEL;
    const float* __restrict__ b1a = b1 + (size_t)asset * HIDDEN;
    const float* __restrict__ W2a = W2 + (size_t)asset * HIDDEN;   // W2 is [N,1,H]

    // A-fragment addressing for 16x4 f32 A tile:
    //   lanes 0-15 hold K={k0,k0+1}, lanes 16-31 hold K={k0+2,k0+3}, row = lane&15
    const int arow  = lane & 15;          // A row within tile; also B column (sample slot)
    const int khalf = (lane >> 4) << 1;   // 0 or 2

    for (int c = 0; c < cnt; c += 16) {
        const int nc = (cnt - c) < 16 ? (cnt - c) : 16;

        // ---- stage up to 16 h vectors (zero unused columns) ----
        for (int r = 0; r < 16; ++r) {
            float v = 0.0f;
            if (r < nc) v = h[(size_t)s_idx[c + r] * D_MODEL + tid]; // tid in [0,256): coalesced
            s_h[r][tid] = v;
        }
        __syncthreads();

        float partial = 0.0f;

        // ---- 8 tiles of 16 hidden rows per wave ----
        for (int j = 0; j < 8; ++j) {
            const int rowbase = (wave * 8 + j) * 16;
            const float* ap = W1a + (size_t)(rowbase + arow) * D_MODEL + khalf;
            v8f acc = {};
            #pragma unroll 8
            for (int k0 = 0; k0 < D_MODEL; k0 += 4) {
                v2f a = *(const v2f*)(ap + k0);                 // global_load_b64
                v2f b = *(const v2f*)(&s_h[arow][k0 + khalf]);  // ds_load_b64, bank-conflict-free
                acc = __builtin_amdgcn_wmma_f32_16x16x4_f32(
                    /*neg_a=*/false, a, /*neg_b=*/false, b,
                    /*c_mod=*/(short)0, acc, /*reuse_a=*/false, /*reuse_b=*/false);
            }
            // D layout: VGPR j -> M = j + 8*(lane>>4), N = lane&15 (= sample column)
            const int mofs = (lane >> 4) << 3;
            #pragma unroll
            for (int jj = 0; jj < 8; ++jj) {
                const int row = rowbase + mofs + jj;
                float z = acc[jj] + b1a[row];
                z = z > 0.0f ? z : 0.0f;
                partial = fmaf(z, W2a[row], partial);
            }
        }

        // ---- cross-wave reduction: lanes n and n+16 hold complementary row halves ----
        s_red[wave][lane] = partial;
        __syncthreads();
        if (tid < 16) {
            float tot = 0.0f;
            #pragma unroll
            for (int w = 0; w < 8; ++w) tot += s_red[w][tid] + s_red[w][tid + 16];
            if (tid < nc) out[s_idx[c + tid]] = tot + b2[asset];
        }
        __syncthreads();
    }
}

extern "C" void kernel_launch(void* const* d_in, const int* in_sizes, int n_in,
                              void* d_out, int out_size, void* d_ws, size_t ws_size,
                              hipStream_t stream) {
    (void)in_sizes; (void)n_in; (void)d_ws; (void)ws_size; (void)out_size;
    const float* h   = (const float*)d_in[0];
    const int*   ids = (const int*)  d_in[1];
    const float* W1  = (const float*)d_in[2];
    const float* b1  = (const float*)d_in[3];
    const float* W2  = (const float*)d_in[4];
    const float* b2  = (const float*)d_in[5];
    float* out = (float*)d_out;
    asset_mlp_wmma_kernel<<<N_ASSETS, 256, 0, stream>>>(h, ids, W1, b1, W2, b2, out);
}